// myGAT_4784593568260
// MI455X (gfx1250) — compile-verified
//
#include <hip/hip_runtime.h>

#define NNODES 50000
#define NEDGES 800000
#define INFEAT 512
#define NHEAD  8
#define DHEAD  64
#define HD     512          // NHEAD * DHEAD
#define NEG_SLOPE 0.2f
#define LDSTRIDE 516        // 512 + 4 pad floats -> conflict-free LDS strides

typedef float v2f __attribute__((ext_vector_type(2)));
typedef float v8f __attribute__((ext_vector_type(8)));

__device__ __forceinline__ float leaky(float x) { return x > 0.f ? x : NEG_SLOPE * x; }

// monotone float <-> uint mapping so integer atomicMax == float max
__device__ __forceinline__ unsigned fenc(float x) {
    unsigned u = __float_as_uint(x);
    return (u & 0x80000000u) ? ~u : (u | 0x80000000u);
}
__device__ __forceinline__ float fdec(unsigned u) {
    return (u & 0x80000000u) ? __uint_as_float(u & 0x7fffffffu) : __uint_as_float(~u);
}

__device__ __forceinline__ void atomAddF(float* p, float v) {
    __hip_atomic_fetch_add(p, v, __ATOMIC_RELAXED, __HIP_MEMORY_SCOPE_AGENT);
}

// ---------------------------------------------------------------------------
// Fused: ft = feat@Wfc, el/er per-head attention dots, out = feat@Wres.
// grid.x = NNODES/16 (N divides exactly: 3125*16 == 50000, so NO row guards),
// 256 threads (8 waves; wave w <=> head w, owning cols [64w, 64w+64)).
// Single k-loop feeds both accumulator sets from one LDS A-fragment.
// ---------------------------------------------------------------------------
__global__ __launch_bounds__(256) void gat_gemm_fused(
    const float* __restrict__ feat, const float* __restrict__ Wfc,
    const float* __restrict__ Wres, const float* __restrict__ attn_l,
    const float* __restrict__ attn_r,
    float* __restrict__ ft, float* __restrict__ el, float* __restrict__ er,
    float* __restrict__ out)
{
    __shared__ float Alds[16 * LDSTRIDE];
    const int m0  = blockIdx.x * 16;
    const int tid = threadIdx.x;

    // cooperative float4 load of 16x512 A strip into padded LDS (no guards:
    // every row index is < NNODES by construction)
    #pragma unroll
    for (int j = 0; j < 8; ++j) {
        int f4  = j * 256 + tid;            // 0..2047 float4 slots
        int row = f4 >> 7;                  // 128 float4 per row
        int k4  = (f4 & 127) << 2;
        float4 v = *(const float4*)(feat + (size_t)(m0 + row) * INFEAT + k4);
        Alds[row * LDSTRIDE + k4 + 0] = v.x;
        Alds[row * LDSTRIDE + k4 + 1] = v.y;
        Alds[row * LDSTRIDE + k4 + 2] = v.z;
        Alds[row * LDSTRIDE + k4 + 3] = v.w;
    }
    __syncthreads();

    const int lane = tid & 31;
    const int wv   = tid >> 5;              // wave id == head id
    const int col  = lane & 15;             // N index within 16-wide tile
    const int hi   = lane >> 4;             // half-wave select
    const int mrow = lane & 15;             // A-matrix M index for this lane

    // fused k-loop: one A fragment feeds 8 WMMAs (4 tiles x {Wfc, Wres})
    v8f accF[4] = {}, accR[4] = {};
    for (int k = 0; k < INFEAT; k += 4) {
        v2f a;
        a.x = Alds[mrow * LDSTRIDE + k + 2 * hi];
        a.y = Alds[mrow * LDSTRIDE + k + 2 * hi + 1];
        #pragma unroll
        for (int t = 0; t < 4; ++t) {
            const int gcol = wv * 64 + t * 16 + col;
            const size_t r0 = (size_t)(k + 2 * hi) * HD + gcol;
            v2f bF, bR;
            bF.x = Wfc [r0];
            bF.y = Wfc [r0 + HD];
            bR.x = Wres[r0];
            bR.y = Wres[r0 + HD];
            accF[t] = __builtin_amdgcn_wmma_f32_16x16x4_f32(
                false, a, false, bF, (short)0, accF[t], false, false);
            accR[t] = __builtin_amdgcn_wmma_f32_16x16x4_f32(
                false, a, false, bR, (short)0, accR[t], false, false);
        }
    }

    // straight-line stores (row = i + 8*hi per C/D layout); no guards needed
    #pragma unroll
    for (int t = 0; t < 4; ++t) {
        const int gcol = wv * 64 + t * 16 + col;
        #pragma unroll
        for (int i = 0; i < 8; ++i) {
            const size_t gm = (size_t)(m0 + i + 8 * hi);
            ft [gm * HD + gcol] = accF[t][i];
            out[gm * HD + gcol] = accR[t][i];
        }
    }

    // el/er: full per-head dot — reduce 64 cols (this wave owns the head)
    #pragma unroll
    for (int i = 0; i < 8; ++i) {
        float sl = 0.f, sr = 0.f;
        #pragma unroll
        for (int t = 0; t < 4; ++t) {
            const int cc = wv * 64 + t * 16 + col;
            float v = accF[t][i];
            sl += v * attn_l[cc];
            sr += v * attn_r[cc];
        }
        sl += __shfl_xor(sl, 1, 32);  sr += __shfl_xor(sr, 1, 32);
        sl += __shfl_xor(sl, 2, 32);  sr += __shfl_xor(sr, 2, 32);
        sl += __shfl_xor(sl, 4, 32);  sr += __shfl_xor(sr, 4, 32);
        sl += __shfl_xor(sl, 8, 32);  sr += __shfl_xor(sr, 8, 32);
        if (col == 0) {
            const int gm = m0 + i + 8 * hi;
            el[gm * NHEAD + wv] = sl;
            er[gm * NHEAD + wv] = sr;
        }
    }
}

// ---------------------------------------------------------------------------
// Edge softmax passes (one thread per (edge, head))
// ---------------------------------------------------------------------------
__global__ __launch_bounds__(256) void edge_max(
    const int* __restrict__ src, const int* __restrict__ dst,
    const float* __restrict__ el, const float* __restrict__ er,
    unsigned* __restrict__ mkey)
{
    int idx = blockIdx.x * 256 + threadIdx.x;
    int e = idx >> 3, h = idx & 7;
    int s = src[e], d = dst[e];
    float x = leaky(el[s * NHEAD + h] + er[d * NHEAD + h]);
    atomicMax(&mkey[d * NHEAD + h], fenc(x));
}

__global__ __launch_bounds__(256) void edge_sum(
    const int* __restrict__ src, const int* __restrict__ dst,
    const float* __restrict__ el, const float* __restrict__ er,
    const unsigned* __restrict__ mkey, float* __restrict__ ssum)
{
    int idx = blockIdx.x * 256 + threadIdx.x;
    int e = idx >> 3, h = idx & 7;
    int s = src[e], d = dst[e];
    float x = leaky(el[s * NHEAD + h] + er[d * NHEAD + h]);
    float m = fdec(mkey[d * NHEAD + h]);
    atomAddF(&ssum[d * NHEAD + h], expf(x - m));
}

__global__ __launch_bounds__(256) void edge_coef(
    const int* __restrict__ src, const int* __restrict__ dst,
    const float* __restrict__ el, const float* __restrict__ er,
    const unsigned* __restrict__ mkey, const float* __restrict__ ssum,
    float* __restrict__ acoef)
{
    int idx = blockIdx.x * 256 + threadIdx.x;
    int e = idx >> 3, h = idx & 7;
    int s = src[e], d = dst[e];
    float x = leaky(el[s * NHEAD + h] + er[d * NHEAD + h]);
    float m = fdec(mkey[d * NHEAD + h]);
    float sm = fmaxf(ssum[d * NHEAD + h], 1e-9f);
    acoef[idx] = expf(x - m) / sm;
}

// ---------------------------------------------------------------------------
// Weighted scatter-add: one thread per (edge, float4 column)
// ---------------------------------------------------------------------------
__global__ __launch_bounds__(256) void edge_scatter(
    const int* __restrict__ src, const int* __restrict__ dst,
    const float* __restrict__ acoef, const float* __restrict__ ft,
    float* __restrict__ out)
{
    int idx = blockIdx.x * 256 + threadIdx.x;   // < NEDGES * 128
    int e  = idx >> 7;
    int q  = idx & 127;
    int c4 = q << 2;            // column (multiple of 4)
    int h  = q >> 4;            // head = c4 / 64
    int s = src[e], d = dst[e];
    float av = acoef[e * NHEAD + h];
    float4 f = *(const float4*)(ft + (size_t)s * HD + c4);
    float* o = out + (size_t)d * HD + c4;
    atomAddF(o + 0, f.x * av);
    atomAddF(o + 1, f.y * av);
    atomAddF(o + 2, f.z * av);
    atomAddF(o + 3, f.w * av);
}

// ---------------------------------------------------------------------------
extern "C" void kernel_launch(void* const* d_in, const int* in_sizes, int n_in,
                              void* d_out, int out_size, void* d_ws, size_t ws_size,
                              hipStream_t stream)
{
    const float* feat   = (const float*)d_in[0];
    const int*   src    = (const int*)  d_in[1];
    const int*   dst    = (const int*)  d_in[2];
    const float* Wfc    = (const float*)d_in[3];
    const float* Wres   = (const float*)d_in[4];
    const float* attn_l = (const float*)d_in[5];
    const float* attn_r = (const float*)d_in[6];
    float* out = (float*)d_out;

    // workspace layout
    float*    ft    = (float*)d_ws;                               // N*512
    float*    el    = ft + (size_t)NNODES * HD;                   // N*8
    float*    er    = el + (size_t)NNODES * NHEAD;                // N*8
    unsigned* mkey  = (unsigned*)(er + (size_t)NNODES * NHEAD);   // N*8
    float*    ssum  = (float*)(mkey + (size_t)NNODES * NHEAD);    // N*8
    float*    acoef = ssum + (size_t)NNODES * NHEAD;              // E*8

    // zero max-keys (key 0 == -inf sentinel) and softmax sums (adjacent)
    hipMemsetAsync(mkey, 0, sizeof(unsigned) * (size_t)NNODES * NHEAD * 2, stream);

    gat_gemm_fused<<<NNODES / 16, 256, 0, stream>>>(
        feat, Wfc, Wres, attn_l, attn_r, ft, el, er, out);

    const int ehBlocks = (NEDGES * NHEAD) / 256;   // 25000 (exact)
    edge_max <<<ehBlocks, 256, 0, stream>>>(src, dst, el, er, mkey);
    edge_sum <<<ehBlocks, 256, 0, stream>>>(src, dst, el, er, mkey, ssum);
    edge_coef<<<ehBlocks, 256, 0, stream>>>(src, dst, el, er, mkey, ssum, acoef);

    const int scBlocks = (NEDGES * 128) / 256;     // 400000 (exact)
    edge_scatter<<<scBlocks, 256, 0, stream>>>(src, dst, acoef, ft, out);
}